// SSIMLoss_44693429682739
// MI455X (gfx1250) — compile-verified
//
#include <hip/hip_runtime.h>

typedef float v2f __attribute__((ext_vector_type(2)));
typedef float v8f __attribute__((ext_vector_type(8)));

namespace {
constexpr int IMG    = 512;
constexpr int PLANES = 48;          // 16 batch * 3 channels
constexpr int BTX    = 64;          // block tile width  (4 waves across)
constexpr int BTY    = 32;          // block tile height (2 waves down)
constexpr int PROWS  = 48;          // BTY + 16 rows (halo + chunk coverage)
constexpr int PPITCH = 76;          // 64 + 10 halo, padded even; cols 74/75 zero (K-pad)
constexpr int HPITCH = 34;          // per-column pitch of col-major H scratch (32 rows)
constexpr float C1v  = 0.0001f;     // 0.01^2
constexpr float C2v  = 0.0009f;     // 0.03^2
constexpr float INVN = 1.0f / 12582912.0f;  // 16*3*512*512
}

// 11-tap Gaussian (sigma=1.5), normalized; returns w[d] for d in [0,10], else 0.
__device__ __forceinline__ float band_w(int d) {
  float v = 0.0f;
  v = (d == 0 || d == 10) ? 0.00102838f : v;
  v = (d == 1 || d ==  9) ? 0.00759863f : v;
  v = (d == 2 || d ==  8) ? 0.03600077f : v;
  v = (d == 3 || d ==  7) ? 0.10936017f : v;
  v = (d == 4 || d ==  6) ? 0.21300553f : v;
  v = (d == 5)            ? 0.26601335f : v;
  return v;
}

__global__ __launch_bounds__(256)
void ssim_wmma_kernel(const float* __restrict__ pred,
                      const float* __restrict__ targ,
                      float* __restrict__ partials) {
  __shared__ __align__(16) float sP[PROWS * PPITCH];
  __shared__ __align__(16) float sT[PROWS * PPITCH];
  __shared__ __align__(16) float sH[8][16 * HPITCH];   // per-wave, one signal at a time
  __shared__ float sRed[256];

  const int tid   = threadIdx.x;
  const int tileX = blockIdx.x * BTX;
  const int tileY = blockIdx.y * BTY;
  const float* pP = pred + (size_t)blockIdx.z * IMG * IMG;
  const float* pT = targ + (size_t)blockIdx.z * IMG * IMG;

  // ---- stage zero-padded pred/target patches into LDS (reference uses zero pad) ----
  for (int idx = tid; idx < PROWS * PPITCH; idx += 256) {
    int row = idx / PPITCH;
    int col = idx - row * PPITCH;
    int gy  = tileY - 5 + row;
    int gx  = tileX - 5 + col;
    bool inb = (col < 74) && ((unsigned)gy < (unsigned)IMG) && ((unsigned)gx < (unsigned)IMG);
    int g = gy * IMG + gx;
    sP[idx] = inb ? pP[g] : 0.0f;
    sT[idx] = inb ? pT[g] : 0.0f;
  }
  __syncthreads();

  const int wave = tid >> 5;
  const int lane = tid & 31;
  const int l16  = lane & 15;
  const int khal = lane >> 4;      // which K/M half of the wave
  const int cx   = wave & 3;       // tile x within block
  const int ry   = wave >> 2;      // tile y within block

  // Banded Gaussian weight operands. Wh = Wv^T, and the WMMA A-layout of X equals
  // the B-layout of X^T, so ONE register set serves both passes:
  //   horizontal: B_t = Wh rows [4t..4t+3]      (index i = column n)
  //   vertical:   A_t = Wv cols [4t..4t+3]      (index i = row m)
  v2f wreg[7];
#pragma unroll
  for (int t = 0; t < 7; ++t) {
    int r = 4 * t + 2 * khal;
    wreg[t][0] = band_w(r - l16);
    wreg[t][1] = band_w(r + 1 - l16);
  }

  float* hbuf = sH[wave];
  v8f acc[5];

#pragma unroll
  for (int sig = 0; sig < 5; ++sig) {
    // ---- horizontal pass: two 16-row chunks, K = 28 (7 x WMMA f32 16x16x4) ----
#pragma unroll
    for (int c = 0; c < 2; ++c) {
      v8f h = {0.f, 0.f, 0.f, 0.f, 0.f, 0.f, 0.f, 0.f};
#pragma unroll
      for (int t = 0; t < 7; ++t) {
        int prow = 16 * ry + 16 * c + l16;
        int pcol = 16 * cx + 4 * t + 2 * khal;
        int off  = prow * PPITCH + pcol;
        v2f a;
        if (sig == 0)      { a = *(const v2f*)&sP[off]; }
        else if (sig == 1) { a = *(const v2f*)&sT[off]; }
        else if (sig == 2) { v2f x = *(const v2f*)&sP[off]; a = x * x; }
        else if (sig == 3) { v2f x = *(const v2f*)&sT[off]; a = x * x; }
        else               { v2f x = *(const v2f*)&sP[off];
                             v2f y = *(const v2f*)&sT[off]; a = x * y; }
        h = __builtin_amdgcn_wmma_f32_16x16x4_f32(false, a, false, wreg[t],
                                                  (short)0, h, false, false);
      }
      // C-layout rows are contiguous per lane-half: 4 x ds_store_b64 into
      // column-major H scratch (row = 16c + 8*khal + v, col = l16).
#pragma unroll
      for (int p = 0; p < 4; ++p) {
        v2f st; st[0] = h[2 * p]; st[1] = h[2 * p + 1];
        *(v2f*)&hbuf[l16 * HPITCH + 16 * c + 8 * khal + 2 * p] = st;
      }
    }
    asm volatile("s_wait_dscnt 0" ::: "memory");
    // ---- vertical pass: O = Wv(16x28) x H(28x16), 7 x WMMA ----
    v8f o = {0.f, 0.f, 0.f, 0.f, 0.f, 0.f, 0.f, 0.f};
#pragma unroll
    for (int t = 0; t < 7; ++t) {
      v2f b = *(const v2f*)&hbuf[l16 * HPITCH + 4 * t + 2 * khal];
      o = __builtin_amdgcn_wmma_f32_16x16x4_f32(false, wreg[t], false, b,
                                                (short)0, o, false, false);
    }
    acc[sig] = o;
  }

  // ---- per-pixel SSIM from the five blurred maps, then block reduction ----
  float psum = 0.0f;
#pragma unroll
  for (int e = 0; e < 8; ++e) {
    float mx  = acc[0][e], my = acc[1][e];
    float xx  = acc[2][e], yy = acc[3][e], xy = acc[4][e];
    float mx2 = mx * mx, my2 = my * my, mxy = mx * my;
    float sx  = xx - mx2, sy = yy - my2, sxy = xy - mxy;
    float num = (2.0f * mxy + C1v) * (2.0f * sxy + C2v);
    float den = (mx2 + my2 + C1v) * (sx + sy + C2v);
    psum += num / den;
  }
  sRed[tid] = psum;
  __syncthreads();
#pragma unroll
  for (int s = 128; s > 0; s >>= 1) {
    if (tid < s) sRed[tid] += sRed[tid + s];
    __syncthreads();
  }
  if (tid == 0) {
    int bid = (blockIdx.z * gridDim.y + blockIdx.y) * gridDim.x + blockIdx.x;
    partials[bid] = sRed[0];
  }
}

__global__ __launch_bounds__(256)
void ssim_finalize_kernel(const float* __restrict__ partials, int n,
                          float* __restrict__ out) {
  __shared__ float red[256];
  float s = 0.0f;
  for (int i = threadIdx.x; i < n; i += 256) s += partials[i];
  red[threadIdx.x] = s;
  __syncthreads();
#pragma unroll
  for (int k = 128; k > 0; k >>= 1) {
    if (threadIdx.x < k) red[threadIdx.x] += red[threadIdx.x + k];
    __syncthreads();
  }
  if (threadIdx.x == 0) out[0] = 1.0f - red[0] * INVN;
}

extern "C" void kernel_launch(void* const* d_in, const int* in_sizes, int n_in,
                              void* d_out, int out_size, void* d_ws, size_t ws_size,
                              hipStream_t stream) {
  const float* pred = (const float*)d_in[0];
  const float* targ = (const float*)d_in[1];
  float* partials = (float*)d_ws;

  dim3 grid(IMG / BTX, IMG / BTY, PLANES);   // 8 x 16 x 48 = 6144 blocks
  ssim_wmma_kernel<<<grid, dim3(256), 0, stream>>>(pred, targ, partials);

  int nPart = grid.x * grid.y * grid.z;
  ssim_finalize_kernel<<<1, dim3(256), 0, stream>>>(partials, nPart, (float*)d_out);
}